// DCGRL_32856499814845
// MI455X (gfx1250) — compile-verified
//
#include <hip/hip_runtime.h>

typedef float v2f __attribute__((ext_vector_type(2)));
typedef float v8f __attribute__((ext_vector_type(8)));

static constexpr int B_   = 8;
static constexpr int N_   = 1024;
static constexpr int KNN_ = 20;
static constexpr int M_   = 256;   // N/4

#define M_PLAIN 0
#define M_EDGE  1
#define M_LPFA  2
#define M_GDIFF 3

#if defined(__HIP_DEVICE_COMPILE__) && __has_builtin(__builtin_amdgcn_wmma_f32_16x16x4_f32)
#define USE_WMMA_F32 1
#else
#define USE_WMMA_F32 0
#endif

// ----------------------------------------------------------------- A operand
// Per-row setup: resolve the gather once; inner loop is branchless.
template <int MODE>
__device__ __forceinline__ void setup_row(const float* __restrict__ X,
                                          const int* __restrict__ idx,
                                          int r, int K, int Npts, int Ci,
                                          const float*& pm, const float*& pn) {
  if (MODE == M_PLAIN) {
    pm = X + (size_t)r * K;
    pn = pm;
  } else {
    int j  = r % KNN_;
    int bn = r / KNN_;
    int n  = bn % Npts;
    int b  = bn / Npts;
    int m  = idx[(size_t)bn * KNN_ + j];
    int cc = (MODE == M_LPFA) ? 3 : Ci;
    const float* xb = X + (size_t)b * Npts * cc;
    pm = xb + (size_t)m * cc;
    pn = xb + (size_t)n * cc;
  }
}

// Branchless A element at (already clamped) k.
template <int MODE>
__device__ __forceinline__ float aval(const float* __restrict__ pm,
                                      const float* __restrict__ pn, int k, int Ci) {
  if (MODE == M_PLAIN) return pm[k];
  if (MODE == M_GDIFF) return pm[k] - pn[k];
  if (MODE == M_EDGE) {                       // [nb - x | nb]
    int kk = (k < Ci) ? k : (k - Ci);
    float vm = pm[kk], vn = pn[kk];
    return (k < Ci) ? (vm - vn) : vm;
  }
  // M_LPFA: [x | nb - x | nb], Ci==3
  int kk = (k < 6) ? ((k < 3) ? k : k - 3) : k - 6;
  float vm = pm[kk], vn = pn[kk];
  return (k < 3) ? vn : ((k < 6) ? (vm - vn) : vm);
}

// ------------------------------------------------------------- WMMA GEMM
// Y[r, o] = sum_k A(r,k) * W[o, k] (+bias[o]); W row-major (Co, K).
// One wave computes a 32x32 macro-tile = 2x2 V_WMMA_F32_16X16X4_F32 tiles.
// Out-of-range rows/cols are CLAMPED (computed but never stored); K tail is
// handled by zeroing only the B element -> no divergence, EXEC stays all-1s.
template <int MODE>
__global__ __launch_bounds__(32) void gemm16_kernel(
    const float* __restrict__ X, const float* __restrict__ W,
    const float* __restrict__ bias, float* __restrict__ Y,
    const int* __restrict__ idx, int R, int K, int Co, int Npts, int Ci) {
  int lane = threadIdx.x;
  int l15  = lane & 15;
  int kh   = lane >> 4;  // 0 or 1

  int r0 = blockIdx.x * 32 + l15;
  int r1 = r0 + 16;
  int c0 = blockIdx.y * 32 + l15;
  int c1 = c0 + 16;
  int r0c = r0 < R ? r0 : R - 1;
  int r1c = r1 < R ? r1 : R - 1;
  int c0c = c0 < Co ? c0 : Co - 1;
  int c1c = c1 < Co ? c1 : Co - 1;

  const float *pm0, *pn0, *pm1, *pn1;
  setup_row<MODE>(X, idx, r0c, K, Npts, Ci, pm0, pn0);
  setup_row<MODE>(X, idx, r1c, K, Npts, Ci, pm1, pn1);
  const float* Wc0 = W + (size_t)c0c * K;
  const float* Wc1 = W + (size_t)c1c * K;

  v8f acc00 = {0.f,0.f,0.f,0.f,0.f,0.f,0.f,0.f};
  v8f acc01 = acc00, acc10 = acc00, acc11 = acc00;
  int Km1 = K - 1;

#if USE_WMMA_F32
  for (int k0 = 0; k0 < K; k0 += 4) {
    int ka  = k0 + 2 * kh;            // this lane's two K slots
    int kb  = ka + 1;
    int kac = ka < Km1 ? ka : Km1;    // clamped (tail-safe) load indices
    int kbc = kb < Km1 ? kb : Km1;
    v2f a0, a1, b0, b1;
    a0.x = aval<MODE>(pm0, pn0, kac, Ci);
    a0.y = aval<MODE>(pm0, pn0, kbc, Ci);
    a1.x = aval<MODE>(pm1, pn1, kac, Ci);
    a1.y = aval<MODE>(pm1, pn1, kbc, Ci);
    float w00 = Wc0[kac], w01 = Wc0[kbc];
    float w10 = Wc1[kac], w11 = Wc1[kbc];
    b0.x = (ka <= Km1) ? w00 : 0.f;
    b0.y = (kb <= Km1) ? w01 : 0.f;
    b1.x = (ka <= Km1) ? w10 : 0.f;
    b1.y = (kb <= Km1) ? w11 : 0.f;
    acc00 = __builtin_amdgcn_wmma_f32_16x16x4_f32(false, a0, false, b0, (short)0, acc00, false, false);
    acc01 = __builtin_amdgcn_wmma_f32_16x16x4_f32(false, a0, false, b1, (short)0, acc01, false, false);
    acc10 = __builtin_amdgcn_wmma_f32_16x16x4_f32(false, a1, false, b0, (short)0, acc10, false, false);
    acc11 = __builtin_amdgcn_wmma_f32_16x16x4_f32(false, a1, false, b1, (short)0, acc11, false, false);
  }
#else
  for (int k = 0; k < K; k++) {
    float w0 = Wc0[k], w1 = Wc1[k];
#pragma unroll
    for (int i = 0; i < 8; i++) {
      int rr0 = blockIdx.x * 32 + i + 8 * kh;
      int rr1 = rr0 + 16;
      rr0 = rr0 < R ? rr0 : R - 1;
      rr1 = rr1 < R ? rr1 : R - 1;
      const float *qm0, *qn0, *qm1, *qn1;
      setup_row<MODE>(X, idx, rr0, K, Npts, Ci, qm0, qn0);
      setup_row<MODE>(X, idx, rr1, K, Npts, Ci, qm1, qn1);
      float av0 = aval<MODE>(qm0, qn0, k, Ci);
      float av1 = aval<MODE>(qm1, qn1, k, Ci);
      acc00[i] += av0 * w0; acc01[i] += av0 * w1;
      acc10[i] += av1 * w0; acc11[i] += av1 * w1;
    }
  }
#endif

  // Store (guards only here, outside any WMMA)
  float bv0 = (bias && c0 < Co) ? bias[c0] : 0.0f;
  float bv1 = (bias && c1 < Co) ? bias[c1] : 0.0f;
#pragma unroll
  for (int i = 0; i < 8; i++) {
    int rr = blockIdx.x * 32 + i + 8 * kh;   // C/D layout: VGPR i -> M=i (+8 hi lanes)
    if (rr < R) {
      if (c0 < Co) Y[(size_t)rr * Co + c0] = acc00[i] + bv0;
      if (c1 < Co) Y[(size_t)rr * Co + c1] = acc01[i] + bv1;
    }
    int rr2 = rr + 16;
    if (rr2 < R) {
      if (c0 < Co) Y[(size_t)rr2 * Co + c0] = acc10[i] + bv0;
      if (c1 < Co) Y[(size_t)rr2 * Co + c1] = acc11[i] + bv1;
    }
  }
}

// ---------------------------------------------------------------- kNN
__global__ void knn_kernel(const float* __restrict__ x, int* __restrict__ idx,
                           int B, int N, int C) {
  int t = blockIdx.x * blockDim.x + threadIdx.x;
  if (t >= B * N) return;
  int b = t / N, n = t % N;
  const float* xb = x + (size_t)b * N * C;
  float bd[KNN_];
  int   bi[KNN_];
#pragma unroll
  for (int j = 0; j < KNN_; j++) { bd[j] = 3.4e38f; bi[j] = 0; }
  for (int m = 0; m < N; m++) {
    float d = 0.0f;
    for (int c = 0; c < C; c++) {
      float df = xb[(size_t)n * C + c] - xb[(size_t)m * C + c];
      d += df * df;
    }
    if (d < bd[KNN_ - 1]) {
      int p = KNN_ - 1;
      while (p > 0 && bd[p - 1] > d) { bd[p] = bd[p - 1]; bi[p] = bi[p - 1]; p--; }
      bd[p] = d; bi[p] = m;
    }
  }
  for (int j = 0; j < KNN_; j++) idx[(size_t)t * KNN_ + j] = bi[j];
}

// ------------------------------------------------------------- BN stats + apply
__global__ void stats_kernel(const float* __restrict__ Y, int R, int C,
                             float* __restrict__ mean, float* __restrict__ rstd) {
  int c = blockIdx.x;
  __shared__ float ssum[256], ssq[256];
  float s = 0.f, q = 0.f;
  for (int r = threadIdx.x; r < R; r += blockDim.x) {
    float v = Y[(size_t)r * C + c];
    s += v; q += v * v;
  }
  ssum[threadIdx.x] = s; ssq[threadIdx.x] = q;
  __syncthreads();
  for (int o = 128; o > 0; o >>= 1) {
    if ((int)threadIdx.x < o) {
      ssum[threadIdx.x] += ssum[threadIdx.x + o];
      ssq[threadIdx.x]  += ssq[threadIdx.x + o];
    }
    __syncthreads();
  }
  if (threadIdx.x == 0) {
    float m = ssum[0] / (float)R;
    float v = ssq[0] / (float)R - m * m;
    mean[c] = m;
    rstd[c] = rsqrtf(v + 1e-5f);
  }
}

__global__ void bnact_kernel(float* __restrict__ Y, long long n, int C,
                             const float* __restrict__ mean,
                             const float* __restrict__ rstd, int act) {
  long long i = (long long)blockIdx.x * blockDim.x + threadIdx.x;
  if (i >= n) return;
  int c = (int)(i % C);
  float v = (Y[i] - mean[c]) * rstd[c];
  if (act == 1) v = v > 0.f ? v : 0.f;
  else if (act == 2) v = v > 0.f ? v : 0.2f * v;
  Y[i] = v;
}

// ------------------------------------------------------------- pools / misc
__global__ void maxpool_k_kernel(const float* __restrict__ Y, float* __restrict__ out,
                                 long long BN, int C) {
  long long i = (long long)blockIdx.x * blockDim.x + threadIdx.x;
  if (i >= BN * C) return;
  long long r = i / C; int c = (int)(i % C);
  const float* p = Y + (r * KNN_) * (long long)C + c;
  float m = p[0];
  for (int j = 1; j < KNN_; j++) m = fmaxf(m, p[(long long)j * C]);
  out[i] = m;
}

__global__ void score_kernel(const float* __restrict__ x, const int* __restrict__ idx,
                             float* __restrict__ score, int B, int N, int C) {
  int t = blockIdx.x * blockDim.x + threadIdx.x;
  if (t >= B * N) return;
  int b = t / N, n = t % N;
  const float* xb = x + (size_t)b * N * C;
  int ids[KNN_];
#pragma unroll
  for (int j = 0; j < KNN_; j++) ids[j] = idx[(size_t)t * KNN_ + j];
  float acc = 0.f;
  for (int c = 0; c < C; c++) {
    float s = 0.f;
    for (int j = 0; j < KNN_; j++) s += xb[(size_t)ids[j] * C + c];
    float d = xb[(size_t)n * C + c] - s * (1.0f / KNN_);
    acc += d * d;
  }
  score[t] = sqrtf(acc);
}

__global__ void topk_select_kernel(const float* __restrict__ score,
                                   int* __restrict__ selTop, int* __restrict__ selBot,
                                   int Nn, int Mm) {
  int b = blockIdx.x;
  __shared__ float sv[1024];
  __shared__ int   si[1024];
  for (int i = threadIdx.x; i < Nn; i += blockDim.x) { sv[i] = score[b * Nn + i]; si[i] = i; }
  __syncthreads();
  for (int ksz = 2; ksz <= Nn; ksz <<= 1) {
    for (int j = ksz >> 1; j > 0; j >>= 1) {
      for (int i = threadIdx.x; i < Nn; i += blockDim.x) {
        int ixj = i ^ j;
        if (ixj > i) {
          float a = sv[i], c = sv[ixj];
          int ai = si[i], ci = si[ixj];
          bool after = (a < c) || (a == c && ai > ci);   // i after ixj in desc order
          bool dirDesc = ((i & ksz) == 0);
          if (dirDesc ? after : !after) { sv[i] = c; sv[ixj] = a; si[i] = ci; si[ixj] = ai; }
        }
      }
      __syncthreads();
    }
  }
  for (int i = threadIdx.x; i < Mm; i += blockDim.x) {
    selTop[b * Mm + i] = si[i];
    selBot[b * Mm + i] = si[Nn - 1 - i];
  }
}

__global__ void gather_rows_kernel(const float* __restrict__ x, const int* __restrict__ sel,
                                   float* __restrict__ out, int B, int N, int M, int C) {
  long long i = (long long)blockIdx.x * blockDim.x + threadIdx.x;
  long long tot = (long long)B * M * C;
  if (i >= tot) return;
  int b = (int)(i / ((long long)M * C));
  long long rem = i % ((long long)M * C);
  int r = (int)(rem / C), c = (int)(rem % C);
  out[i] = x[((size_t)b * N + sel[b * M + r]) * C + c];
}

// ------------------------------------------------------------- fused attention
__global__ void sgcam_kernel(const float* __restrict__ x, const float* __restrict__ q,
                             const float* __restrict__ kk, const float* __restrict__ v,
                             float* __restrict__ out, int B, int N, int M,
                             int ostride, int ooff) {
  __shared__ float sbuf[32 * M_];
  int t = blockIdx.x * 32 + threadIdx.x;
  if (t >= B * N) return;
  int b = t / N;
  const float* qn = q + (size_t)t * 16;
  const float* kb = kk + (size_t)b * M * 16;
  const float* vb = v  + (size_t)b * M * 64;
  float qr[16];
#pragma unroll
  for (int c = 0; c < 16; c++) qr[c] = qn[c];
  float* srow = sbuf + threadIdx.x * M_;
  float mx = -3.4e38f;
  for (int m = 0; m < M; m++) {
    float s = 0.f;
    const float* km = kb + (size_t)m * 16;
#pragma unroll
    for (int c = 0; c < 16; c++) s += qr[c] * km[c];
    srow[m] = s;
    mx = fmaxf(mx, s);
  }
  float denom = 0.f;
  for (int m = 0; m < M; m++) { float e = expf(srow[m] - mx); srow[m] = e; denom += e; }
  float inv = 1.0f / denom;
  float acc[64];
#pragma unroll
  for (int c = 0; c < 64; c++) acc[c] = 0.f;
  for (int m = 0; m < M; m++) {
    float p = srow[m] * inv;
    const float* vm = vb + (size_t)m * 64;
    for (int c = 0; c < 64; c++) acc[c] += p * vm[c];
  }
  float* od = out + (size_t)t * ostride + ooff;
  const float* xn = x + (size_t)t * 64;
  for (int c = 0; c < 64; c++) od[c] = xn[c] + acc[c];
}

// ------------------------------------------------------------- small utils
__global__ void transpose_xyz_kernel(const float* __restrict__ xyz, float* __restrict__ pts,
                                     int B, int N) {
  int i = blockIdx.x * blockDim.x + threadIdx.x;
  if (i >= B * N * 3) return;
  int c = i % 3; int n = (i / 3) % N; int b = i / (3 * N);
  pts[i] = xyz[(size_t)b * 3 * N + (size_t)c * N + n];
}

__global__ void copy_strided_kernel(const float* __restrict__ src, float* __restrict__ dst,
                                    long long rows, int csrc, int cdst, int off) {
  long long i = (long long)blockIdx.x * blockDim.x + threadIdx.x;
  if (i >= rows * csrc) return;
  long long r = i / csrc; int c = (int)(i % csrc);
  dst[r * cdst + off + c] = src[i];
}

__global__ void slice_kernel(const float* __restrict__ src, float* __restrict__ dst,
                             int B, int Nin, int C, int stride) {
  int Nout = Nin / stride;
  long long i = (long long)blockIdx.x * blockDim.x + threadIdx.x;
  long long tot = (long long)B * Nout * C;
  if (i >= tot) return;
  int c = (int)(i % C);
  int n = (int)((i / C) % Nout);
  int b = (int)(i / ((long long)C * Nout));
  dst[i] = src[((size_t)b * Nin + (size_t)n * stride) * C + c];
}

__global__ void addact_kernel(const float* __restrict__ a, const float* __restrict__ b,
                              float* __restrict__ out, long long n, int act) {
  long long i = (long long)blockIdx.x * blockDim.x + threadIdx.x;
  if (i >= n) return;
  float v = a[i] + b[i];
  if (act == 2) v = v > 0.f ? v : 0.2f * v;
  out[i] = v;
}

__global__ void add3_kernel(const float* __restrict__ a, const float* __restrict__ b,
                            const float* __restrict__ c, float* __restrict__ out, long long n) {
  long long i = (long long)blockIdx.x * blockDim.x + threadIdx.x;
  if (i >= n) return;
  out[i] = a[i] + b[i] + c[i];
}

__global__ void pool_points_kernel(const float* __restrict__ z, float* __restrict__ out,
                                   int B, int N, int C) {
  int i = blockIdx.x * blockDim.x + threadIdx.x;
  if (i >= B * C) return;
  int b = i / C, c = i % C;
  const float* p = z + (size_t)b * N * C + c;
  float mx = p[0], s = 0.f;
  for (int n = 0; n < N; n++) { float v = p[(size_t)n * C]; mx = fmaxf(mx, v); s += v; }
  out[(size_t)b * 2 * C + c]     = mx;
  out[(size_t)b * 2 * C + C + c] = s / (float)N;
}

__global__ void layernorm_kernel(float* __restrict__ x, int C) {
  int r = blockIdx.x;
  __shared__ float ss[256], sq[256];
  float s = 0.f, q = 0.f;
  for (int c = threadIdx.x; c < C; c += blockDim.x) {
    float v = x[(size_t)r * C + c];
    s += v; q += v * v;
  }
  ss[threadIdx.x] = s; sq[threadIdx.x] = q;
  __syncthreads();
  for (int o = 128; o > 0; o >>= 1) {
    if ((int)threadIdx.x < o) { ss[threadIdx.x] += ss[threadIdx.x + o]; sq[threadIdx.x] += sq[threadIdx.x + o]; }
    __syncthreads();
  }
  float m  = ss[0] / (float)C;
  float rv = rsqrtf(sq[0] / (float)C - m * m + 1e-5f);
  for (int c = threadIdx.x; c < C; c += blockDim.x)
    x[(size_t)r * C + c] = (x[(size_t)r * C + c] - m) * rv;
}

// ================================================================= host side
static inline unsigned div_up(long long a, long long b) { return (unsigned)((a + b - 1) / b); }

template <int MODE>
static void launch_gemm(const float* X, const float* W, const float* bias, float* Y,
                        const int* idx, int R, int K, int Co, int Npts, int Ci,
                        hipStream_t s) {
  dim3 grid(div_up(R, 32), div_up(Co, 32));
  gemm16_kernel<MODE><<<grid, dim3(32), 0, s>>>(X, W, bias, Y, idx, R, K, Co, Npts, Ci);
}

static void bn_act(float* Y, long long R, int C, int act, float* mean, float* rstd,
                   hipStream_t s) {
  stats_kernel<<<dim3((unsigned)C), dim3(256), 0, s>>>(Y, (int)R, C, mean, rstd);
  long long n = R * (long long)C;
  bnact_kernel<<<dim3(div_up(n, 256)), dim3(256), 0, s>>>(Y, n, C, mean, rstd, act);
}

static void edge_block(const float* x, int Ci, const int* idx, const float* W1,
                       const float* W2, int Co, float* Y1, float* Y2, float* outx,
                       float* mean, float* rstd, hipStream_t s) {
  int R = B_ * N_ * KNN_;
  launch_gemm<M_EDGE>(x, W1, nullptr, Y1, idx, R, 2 * Ci, Co, N_, Ci, s);
  bn_act(Y1, R, Co, 1, mean, rstd, s);
  launch_gemm<M_PLAIN>(Y1, W2, nullptr, Y2, nullptr, R, Co, Co, 0, 0, s);
  bn_act(Y2, R, Co, 1, mean, rstd, s);
  long long bn = (long long)B_ * N_;
  maxpool_k_kernel<<<dim3(div_up(bn * Co, 256)), dim3(256), 0, s>>>(Y2, outx, bn, Co);
}

static void gdm_run(const float* x, int C, int* idxb, float* scoreb, int* selS, int* selG,
                    float* xs, float* xg, hipStream_t s) {
  knn_kernel<<<dim3(div_up(B_ * N_, 128)), dim3(128), 0, s>>>(x, idxb, B_, N_, C);
  score_kernel<<<dim3(div_up(B_ * N_, 128)), dim3(128), 0, s>>>(x, idxb, scoreb, B_, N_, C);
  topk_select_kernel<<<dim3(B_), dim3(512), 0, s>>>(scoreb, selS, selG, N_, M_);
  long long tot = (long long)B_ * M_ * C;
  gather_rows_kernel<<<dim3(div_up(tot, 256)), dim3(256), 0, s>>>(x, selS, xs, B_, N_, M_, C);
  gather_rows_kernel<<<dim3(div_up(tot, 256)), dim3(256), 0, s>>>(x, selG, xg, B_, N_, M_, C);
}

static void sgcam_run(const float* x, const float* y, const float* Wq, const float* Wk,
                      const float* Wv, float* qb, float* kb, float* vb, float* outc,
                      int ooff, hipStream_t s) {
  launch_gemm<M_PLAIN>(x, Wq, nullptr, qb, nullptr, B_ * N_, 64, 16, 0, 0, s);
  launch_gemm<M_PLAIN>(y, Wk, nullptr, kb, nullptr, B_ * M_, 64, 16, 0, 0, s);
  launch_gemm<M_PLAIN>(y, Wv, nullptr, vb, nullptr, B_ * M_, 64, 64, 0, 0, s);
  sgcam_kernel<<<dim3((B_ * N_) / 32), dim3(32), 0, s>>>(x, qb, kb, vb, outc, B_, N_, M_,
                                                         128, ooff);
}

extern "C" void kernel_launch(void* const* d_in, const int* in_sizes, int n_in,
                              void* d_out, int out_size, void* d_ws, size_t ws_size,
                              hipStream_t stream) {
  (void)out_size; (void)ws_size;
  // ---- locate xyz (only input with 8*3*1024 elements), params in dict order
  int xyz_ix = -1;
  for (int i = 0; i < n_in; i++)
    if (in_sizes[i] == B_ * 3 * N_) { xyz_ix = i; break; }
  if (xyz_ix < 0) xyz_ix = 0;
  int pi = 0;
  auto nextp = [&]() -> const float* {
    while (pi < n_in && (pi == xyz_ix || in_sizes[pi] == 0)) pi++;
    const float* p = (pi < n_in) ? (const float*)d_in[pi] : nullptr;
    pi++;
    return p;
  };
  const float* xyz = (const float*)d_in[xyz_ix];

  // gda weights (gb* are dead: BN cancels per-channel bias, but must be consumed)
  const float *gW1 = nextp(); nextp(); const float *gW11 = nextp(); nextp();
  const float *gW12 = nextp(); nextp(); const float *gW2 = nextp(); nextp();
  const float *gW21 = nextp(); nextp(); const float *gW22 = nextp(); nextp();
  const float *gW3 = nextp(); nextp(); const float *gW31 = nextp(); nextp();
  const float *gW32 = nextp(); nextp(); const float *gW4 = nextp(); nextp();
  const float *sg1s_q = nextp(), *sg1s_k = nextp(), *sg1s_v = nextp();
  const float *sg1g_q = nextp(), *sg1g_k = nextp(), *sg1g_v = nextp();
  const float *sg2s_q = nextp(), *sg2s_k = nextp(), *sg2s_v = nextp();
  const float *sg2g_q = nextp(), *sg2g_k = nextp(), *sg2g_v = nextp();
  const float *lpfaW = nextp();
  struct CicW { const float *Win, *Wgrp, *Wout, *Wsc; };
  CicW cw[8];
  const bool hasSc[8] = {true, false, true, false, true, false, true, false};
  for (int i = 0; i < 8; i++) {
    cw[i].Win = nextp(); cw[i].Wgrp = nextp(); cw[i].Wout = nextp();
    cw[i].Wsc = hasSc[i] ? nextp() : nullptr;
  }
  const float *cW0 = nextp();
  const float *Wgp = nextp(), *bgp = nextp();
  const float *Wcp = nextp(), *bcp = nextp();
  const float *Wres = nextp(), *bres = nextp();
  const float *Wfc1 = nextp();
  const float *Wfc2 = nextp(), *bfc2 = nextp();

  // ---- workspace bump allocator
  char* base = (char*)d_ws;
  size_t off = 0;
  auto allocF = [&](size_t n) -> float* {
    size_t a = (off + 255) & ~(size_t)255;
    off = a + n * sizeof(float);
    return (float*)(base + a);
  };
  auto allocI = [&](size_t n) -> int* {
    size_t a = (off + 255) & ~(size_t)255;
    off = a + n * sizeof(int);
    return (int*)(base + a);
  };

  float* pts  = allocF((size_t)B_ * N_ * 3);
  int*   idx0 = allocI((size_t)B_ * N_ * KNN_);
  int*   idxF = allocI((size_t)B_ * N_ * KNN_);
  float* E1   = allocF((size_t)B_ * N_ * KNN_ * 128);
  float* E2   = allocF((size_t)B_ * N_ * KNN_ * 128);
  float* x1   = allocF((size_t)B_ * N_ * 64);
  float* z1   = allocF((size_t)B_ * N_ * 64);
  float* x1t  = allocF((size_t)B_ * N_ * 67);
  float* x2   = allocF((size_t)B_ * N_ * 64);
  float* z2   = allocF((size_t)B_ * N_ * 64);
  float* x2t  = allocF((size_t)B_ * N_ * 131);
  float* x3   = allocF((size_t)B_ * N_ * 128);
  float* z3   = allocF((size_t)B_ * N_ * 128);
  float* zcat = allocF((size_t)B_ * N_ * 256);
  float* zbig = allocF((size_t)B_ * N_ * 512);
  float* gdaO = allocF((size_t)B_ * 1024);
  float* scr  = allocF((size_t)B_ * N_);
  int*   selS = allocI((size_t)B_ * M_);
  int*   selG = allocI((size_t)B_ * M_);
  float* xs   = allocF((size_t)B_ * M_ * 64);
  float* xg   = allocF((size_t)B_ * M_ * 64);
  float* qb   = allocF((size_t)B_ * N_ * 16);
  float* kb   = allocF((size_t)B_ * M_ * 16);
  float* vb   = allocF((size_t)B_ * M_ * 64);
  float* zc1  = allocF((size_t)B_ * N_ * 128);
  float* mean = allocF(1024);
  float* rstd = allocF(1024);
  float* fA   = allocF((size_t)B_ * N_ * 128);
  float* fB   = allocF((size_t)B_ * N_ * 128);
  float* xyzA = allocF((size_t)B_ * N_ * 3);
  float* xyzB = allocF((size_t)B_ * N_ * 3);
  float* npts = allocF((size_t)B_ * N_ * 128);
  float* hbuf = allocF((size_t)B_ * N_ * 64);
  float* h2b  = allocF((size_t)B_ * N_ * 64);
  float* mainb= allocF((size_t)B_ * N_ * 128);
  float* scb  = allocF((size_t)B_ * N_ * 128);
  float* zcb  = allocF((size_t)B_ * 64 * 1024);
  float* curO = allocF((size_t)B_ * 2048);
  float* gvec = allocF((size_t)B_ * 1024);
  float* cvec = allocF((size_t)B_ * 1024);
  float* gc   = allocF((size_t)B_ * 2048);
  float* resv = allocF((size_t)B_ * 1024);
  float* xh   = allocF((size_t)B_ * 1024);
  float* hfc  = allocF((size_t)B_ * 512);

  hipStream_t s = stream;

  // ---- pts = transpose(xyz)
  transpose_xyz_kernel<<<dim3(div_up((long long)B_ * N_ * 3, 256)), dim3(256), 0, s>>>(
      xyz, pts, B_, N_);

  // ================= GDA branch =================
  knn_kernel<<<dim3(div_up(B_ * N_, 128)), dim3(128), 0, s>>>(pts, idx0, B_, N_, 3);
  edge_block(pts, 3, idx0, gW1, gW11, 64, E1, E2, x1, mean, rstd, s);

  gdm_run(x1, 64, idxF, scr, selS, selG, xs, xg, s);
  sgcam_run(x1, xs, sg1s_q, sg1s_k, sg1s_v, qb, kb, vb, zc1, 0, s);
  sgcam_run(x1, xg, sg1g_q, sg1g_k, sg1g_v, qb, kb, vb, zc1, 64, s);
  launch_gemm<M_PLAIN>(zc1, gW12, nullptr, z1, nullptr, B_ * N_, 128, 64, 0, 0, s);
  bn_act(z1, (long long)B_ * N_, 64, 1, mean, rstd, s);

  copy_strided_kernel<<<dim3(div_up((long long)B_ * N_ * 3, 256)), dim3(256), 0, s>>>(
      pts, x1t, (long long)B_ * N_, 3, 67, 0);
  copy_strided_kernel<<<dim3(div_up((long long)B_ * N_ * 64, 256)), dim3(256), 0, s>>>(
      z1, x1t, (long long)B_ * N_, 64, 67, 3);

  knn_kernel<<<dim3(div_up(B_ * N_, 128)), dim3(128), 0, s>>>(x1t, idxF, B_, N_, 67);
  edge_block(x1t, 67, idxF, gW2, gW21, 64, E1, E2, x2, mean, rstd, s);

  gdm_run(x2, 64, idxF, scr, selS, selG, xs, xg, s);
  sgcam_run(x2, xs, sg2s_q, sg2s_k, sg2s_v, qb, kb, vb, zc1, 0, s);
  sgcam_run(x2, xg, sg2g_q, sg2g_k, sg2g_v, qb, kb, vb, zc1, 64, s);
  launch_gemm<M_PLAIN>(zc1, gW22, nullptr, z2, nullptr, B_ * N_, 128, 64, 0, 0, s);
  bn_act(z2, (long long)B_ * N_, 64, 1, mean, rstd, s);

  copy_strided_kernel<<<dim3(div_up((long long)B_ * N_ * 67, 256)), dim3(256), 0, s>>>(
      x1t, x2t, (long long)B_ * N_, 67, 131, 0);
  copy_strided_kernel<<<dim3(div_up((long long)B_ * N_ * 64, 256)), dim3(256), 0, s>>>(
      z2, x2t, (long long)B_ * N_, 64, 131, 67);

  knn_kernel<<<dim3(div_up(B_ * N_, 128)), dim3(128), 0, s>>>(x2t, idxF, B_, N_, 131);
  edge_block(x2t, 131, idxF, gW3, gW31, 128, E1, E2, x3, mean, rstd, s);

  launch_gemm<M_PLAIN>(x3, gW32, nullptr, z3, nullptr, B_ * N_, 128, 128, 0, 0, s);
  bn_act(z3, (long long)B_ * N_, 128, 1, mean, rstd, s);

  copy_strided_kernel<<<dim3(div_up((long long)B_ * N_ * 64, 256)), dim3(256), 0, s>>>(
      z1, zcat, (long long)B_ * N_, 64, 256, 0);
  copy_strided_kernel<<<dim3(div_up((long long)B_ * N_ * 64, 256)), dim3(256), 0, s>>>(
      z2, zcat, (long long)B_ * N_, 64, 256, 64);
  copy_strided_kernel<<<dim3(div_up((long long)B_ * N_ * 128, 256)), dim3(256), 0, s>>>(
      z3, zcat, (long long)B_ * N_, 128, 256, 128);
  launch_gemm<M_PLAIN>(zcat, gW4, nullptr, zbig, nullptr, B_ * N_, 256, 512, 0, 0, s);
  bn_act(zbig, (long long)B_ * N_, 512, 1, mean, rstd, s);
  pool_points_kernel<<<dim3(div_up(B_ * 512, 256)), dim3(256), 0, s>>>(zbig, gdaO, B_, N_, 512);

  // ================= Curve branch =================
  // lpfa: reuses idx0 (kNN over pts)
  launch_gemm<M_LPFA>(pts, lpfaW, nullptr, E1, idx0, B_ * N_ * KNN_, 9, 32, N_, 3, s);
  bn_act(E1, (long long)B_ * N_ * KNN_, 32, 1, mean, rstd, s);
  maxpool_k_kernel<<<dim3(div_up((long long)B_ * N_ * 32, 256)), dim3(256), 0, s>>>(
      E1, fA, (long long)B_ * N_, 32);

  struct Cfg { int npoint, ci, co, cb; };
  const Cfg cfgs[8] = {
      {1024, 32, 64, 32}, {1024, 64, 64, 16}, {1024, 64, 128, 64}, {1024, 128, 128, 32},
      {256, 128, 256, 128}, {256, 256, 256, 64}, {64, 256, 512, 256}, {64, 512, 512, 128}};
  float* fCur = fA;
  float* fNext = fB;
  float* xyzCur = pts;
  float* xyzNext = xyzA;
  int Ncur = N_;
  for (int ci_ = 0; ci_ < 8; ci_++) {
    const Cfg& cf = cfgs[ci_];
    int stride = Ncur / cf.npoint;
    int np = cf.npoint;
    slice_kernel<<<dim3(div_up((long long)B_ * np * 3, 256)), dim3(256), 0, s>>>(
        xyzCur, xyzNext, B_, Ncur, 3, stride);
    slice_kernel<<<dim3(div_up((long long)B_ * np * cf.ci, 256)), dim3(256), 0, s>>>(
        fCur, npts, B_, Ncur, cf.ci, stride);
    // h = lrelu(bn(new_pts @ Win^T))
    launch_gemm<M_PLAIN>(npts, cw[ci_].Win, nullptr, hbuf, nullptr, B_ * np, cf.ci, cf.cb, 0, 0, s);
    bn_act(hbuf, (long long)B_ * np, cf.cb, 2, mean, rstd, s);
    // knn on new_xyz (3ch)
    knn_kernel<<<dim3(div_up(B_ * np, 128)), dim3(128), 0, s>>>(xyzNext, idxF, B_, np, 3);
    // g = gather(h)-h  -> GEMM with Wgrp, lrelu(bn(.)), max over k
    launch_gemm<M_GDIFF>(hbuf, cw[ci_].Wgrp, nullptr, E1, idxF, B_ * np * KNN_, cf.cb, cf.cb,
                         np, cf.cb, s);
    bn_act(E1, (long long)B_ * np * KNN_, cf.cb, 2, mean, rstd, s);
    maxpool_k_kernel<<<dim3(div_up((long long)B_ * np * cf.cb, 256)), dim3(256), 0, s>>>(
        E1, h2b, (long long)B_ * np, cf.cb);
    // main = bn(h2 @ Wout^T)
    launch_gemm<M_PLAIN>(h2b, cw[ci_].Wout, nullptr, mainb, nullptr, B_ * np, cf.cb, cf.co, 0, 0, s);
    bn_act(mainb, (long long)B_ * np, cf.co, 0, mean, rstd, s);
    // shortcut
    const float* scp;
    if (cw[ci_].Wsc) {
      launch_gemm<M_PLAIN>(npts, cw[ci_].Wsc, nullptr, scb, nullptr, B_ * np, cf.ci, cf.co, 0, 0, s);
      bn_act(scb, (long long)B_ * np, cf.co, 0, mean, rstd, s);
      scp = scb;
    } else {
      scp = npts;  // ci == co in these blocks
    }
    long long nn = (long long)B_ * np * cf.co;
    addact_kernel<<<dim3(div_up(nn, 256)), dim3(256), 0, s>>>(mainb, scp, fNext, nn, 2);
    // rotate buffers
    float* t = fCur; fCur = fNext; fNext = (t == pts) ? fB : t;
    if (fNext == fCur) fNext = (fCur == fA) ? fB : fA;
    float* tx = xyzCur; xyzCur = xyzNext; xyzNext = (tx == pts) ? xyzB : tx;
    if (xyzNext == xyzCur) xyzNext = (xyzCur == xyzA) ? xyzB : xyzA;
    Ncur = np;
  }
  // z = lrelu(bn(f @ cW0^T)), pool
  launch_gemm<M_PLAIN>(fCur, cW0, nullptr, zcb, nullptr, B_ * 64, 512, 1024, 0, 0, s);
  bn_act(zcb, (long long)B_ * 64, 1024, 2, mean, rstd, s);
  pool_points_kernel<<<dim3(div_up(B_ * 1024, 256)), dim3(256), 0, s>>>(zcb, curO, B_, 64, 1024);

  // ================= Head =================
  launch_gemm<M_PLAIN>(gdaO, Wgp, bgp, gvec, nullptr, B_, 1024, 1024, 0, 0, s);
  launch_gemm<M_PLAIN>(curO, Wcp, bcp, cvec, nullptr, B_, 2048, 1024, 0, 0, s);
  copy_strided_kernel<<<dim3(div_up((long long)B_ * 1024, 256)), dim3(256), 0, s>>>(
      gvec, gc, B_, 1024, 2048, 0);
  copy_strided_kernel<<<dim3(div_up((long long)B_ * 1024, 256)), dim3(256), 0, s>>>(
      cvec, gc, B_, 1024, 2048, 1024);
  launch_gemm<M_PLAIN>(gc, Wres, bres, resv, nullptr, B_, 2048, 1024, 0, 0, s);
  add3_kernel<<<dim3(div_up((long long)B_ * 1024, 256)), dim3(256), 0, s>>>(
      gvec, cvec, resv, xh, (long long)B_ * 1024);
  layernorm_kernel<<<dim3(B_), dim3(256), 0, s>>>(xh, 1024);
  launch_gemm<M_PLAIN>(xh, Wfc1, nullptr, hfc, nullptr, B_, 1024, 512, 0, 0, s);
  bn_act(hfc, B_, 512, 1, mean, rstd, s);
  launch_gemm<M_PLAIN>(hfc, Wfc2, bfc2, (float*)d_out, nullptr, B_, 512, 40, 0, 0, s);
}